// Mask_61641370632634
// MI455X (gfx1250) — compile-verified
//
#include <hip/hip_runtime.h>

// Elementwise masked-adjacency rebuild for Cora (N=2708), CDNA5/gfx1250.
// out[i,j] = adj[i,j]                          if cluster[i] != cluster[j]
//          = 0                                  if i == j (same cluster)
//          = bernoulli(sigmoid(mask[r,c]))      r=max(i,j), c=min(i,j)
// Bernoulli bits reproduce JAX threefry2x32, key=(0,42), split-counter layout.
//
// Data movement (MI455X-specific):
//  - mask tile staged to LDS by the Tensor Data Mover (tensor_load_to_lds,
//    TENSORcnt), with HW edge-clamp (tensor_dim) and HW bank padding.
//  - cluster-id slices staged via global_load_async_to_lds_b32 (ASYNCcnt).
//  - streamed adj/out use nontemporal b128 (no L2 pollution, no reuse).

#define TS      64
#define LPITCH  68   // 64 + 4 pad: TDM pad_interval=64dw, pad_amount=4dw

typedef float    v4f  __attribute__((ext_vector_type(4)));
typedef unsigned u32x4 __attribute__((ext_vector_type(4)));
typedef unsigned u32x8 __attribute__((ext_vector_type(8)));

__device__ __forceinline__ unsigned rotl32(unsigned x, unsigned r) {
  return (x << r) | (x >> (32u - r));
}

// JAX threefry_2x32 with key (0, 42); flat counter f over n*n elements,
// classic (non-partitionable) scheme: counter = split iota into two halves.
__device__ __forceinline__ unsigned threefry2x32_bits(unsigned f, unsigned half) {
  const bool lo = f < half;
  unsigned x0 = lo ? f : (f - half);
  unsigned x1 = lo ? (f + half) : f;
  const unsigned ks0 = 0u;
  const unsigned ks1 = 42u;
  const unsigned ks2 = 0x1BD11BDAu ^ ks0 ^ ks1;
  x0 += ks0; x1 += ks1;
#define TF_R(r) { x0 += x1; x1 = rotl32(x1, (r)); x1 ^= x0; }
  TF_R(13) TF_R(15) TF_R(26) TF_R(6)
  x0 += ks1; x1 += ks2 + 1u;
  TF_R(17) TF_R(29) TF_R(16) TF_R(24)
  x0 += ks2; x1 += ks0 + 2u;
  TF_R(13) TF_R(15) TF_R(26) TF_R(6)
  x0 += ks0; x1 += ks1 + 3u;
  TF_R(17) TF_R(29) TF_R(16) TF_R(24)
  x0 += ks1; x1 += ks2 + 4u;
  TF_R(13) TF_R(15) TF_R(26) TF_R(6)
  x0 += ks2; x1 += ks0 + 5u;
#undef TF_R
  return lo ? x0 : x1;
}

__launch_bounds__(256)
__global__ void mask_adj_kernel(const float* __restrict__ mask,
                                const float* __restrict__ adj,
                                const int*   __restrict__ cid,
                                float*       __restrict__ out,
                                int n) {
  __shared__ __attribute__((aligned(16))) float mtile[TS][LPITCH];
  __shared__ int scid_r[TS];
  __shared__ int scid_c[TS];

  const int t   = threadIdx.x;
  const int bi0 = blockIdx.x * TS;               // output row block
  const int bj0 = blockIdx.y * TS;               // output col block
  // Mask source tile: the lower-triangle mirror (rows = max block, cols = min)
  const int R0  = (bi0 > bj0) ? bi0 : bj0;
  const int C0  = (bi0 > bj0) ? bj0 : bi0;

  // --- stage cluster ids for this tile's rows/cols via ASYNCcnt path ---
  // waves 0-1: scid_r, waves 2-3: scid_c (one b32 async load per lane)
  if (t < TS) {
    int i = bi0 + t; if (i > n - 1) i = n - 1;
    unsigned la = (unsigned)(unsigned long long)(void*)&scid_r[t];
    unsigned long long ga = (unsigned long long)(const void*)(cid + i);
    asm volatile("global_load_async_to_lds_b32 %0, %1, off"
                 :: "v"(la), "v"(ga) : "memory");
  } else if (t < 2 * TS) {
    int j = bj0 + (t - TS); if (j > n - 1) j = n - 1;
    unsigned la = (unsigned)(unsigned long long)(void*)&scid_c[t - TS];
    unsigned long long ga = (unsigned long long)(const void*)(cid + j);
    asm volatile("global_load_async_to_lds_b32 %0, %1, off"
                 :: "v"(la), "v"(ga) : "memory");
  }

  // --- TDM: one descriptor moves the whole 64x64 mask tile (R0,C0) to LDS ---
  // Issued by wave 0 only (TDM ignores EXEC; one issue per workgroup).
  if (t < 32) {
    unsigned long long ga =
        (unsigned long long)(const void*)(mask + (size_t)R0 * (size_t)n + (size_t)C0);
    unsigned lds_off = (unsigned)(unsigned long long)(void*)&mtile[0][0];
    unsigned td0 = (unsigned)(n - C0);   // remaining cols: OOB reads -> 0
    unsigned td1 = (unsigned)(n - R0);   // remaining rows: OOB reads -> 0

    u32x4 g0;
    g0.x = 1u;                                   // count=1 (valid user D#)
    g0.y = lds_off;                              // lds_addr
    g0.z = (unsigned)ga;                         // global_addr[31:0]
    g0.w = (unsigned)(ga >> 32) | 0x80000000u;   // global_addr[56:32] | type=2

    u32x8 g1;
    g1.s0 = 0x07520000u;  // data_size=4B | pad_enable | pad_interval=64dw | pad_amount=4dw
    g1.s1 = (td0 & 0xFFFFu) << 16;               // tensor_dim0[15:0]
    g1.s2 = (td0 >> 16) | ((td1 & 0xFFFFu) << 16); // tensor_dim0[31:16], tensor_dim1[15:0]
    g1.s3 = (td1 >> 16) | ((unsigned)TS << 16);  // tensor_dim1[31:16], tile_dim0=64
    g1.s4 = (unsigned)TS;                        // tile_dim1=64, tile_dim2=0
    g1.s5 = (unsigned)n;                         // tensor_dim0_stride[31:0]
    g1.s6 = 0u;                                  // stride0[47:32]=0, stride1[15:0]=0
    g1.s7 = 0u;                                  // stride1[47:16]=0

    asm volatile("tensor_load_to_lds %0, %1" :: "s"(g0), "s"(g1) : "memory");
  }

  // Completion: ASYNCcnt (cid slices) + TENSORcnt (mask tile), then publish.
  asm volatile("s_wait_asynccnt 0x0" ::: "memory");
  __builtin_amdgcn_s_wait_tensorcnt(0);
  __syncthreads();

  const unsigned half = (unsigned)n * (unsigned)n / 2u;  // n*n even for n=2708

  // --- compute: each thread produces 4 float4s of the output tile ---
#pragma unroll
  for (int k = 0; k < 4; ++k) {
    int q   = t + k * 256;
    int row = q >> 4;
    int c4  = (q & 15) << 2;
    int i   = bi0 + row;
    int j0  = bj0 + c4;
    if (i >= n) continue;

    const int  ci      = scid_r[row];
    const bool fullvec = (j0 + 3 < n);
    const size_t base  = (size_t)i * (size_t)n + (size_t)j0;

    v4f a4 = {0.f, 0.f, 0.f, 0.f};
    if (fullvec) a4 = __builtin_nontemporal_load((const v4f*)(adj + base));

    v4f res;
#pragma unroll
    for (int e = 0; e < 4; ++e) {
      int j = j0 + e;
      float v = 0.0f;
      if (j < n) {
        int cj = scid_c[c4 + e];
        if (ci != cj) {
          v = fullvec ? a4[e] : adj[base + (size_t)e];
        } else if (i != j) {
          int gr = (i > j) ? i : j;
          int gc = (i > j) ? j : i;
          float m = mtile[gr - R0][gc - C0];
          float p = 1.0f / (1.0f + __expf(-m));
          unsigned bits =
              threefry2x32_bits((unsigned)gr * (unsigned)n + (unsigned)gc, half);
          float u = __uint_as_float((bits >> 9) | 0x3f800000u) - 1.0f;
          v = (u < p) ? 1.0f : 0.0f;
        }
        // same cluster && i==j -> 0 (diagonal excluded by strict lower tri)
      }
      res[e] = v;
    }

    if (fullvec) {
      __builtin_nontemporal_store(res, (v4f*)(out + base));
    } else {
#pragma unroll
      for (int e = 0; e < 4; ++e)
        if (j0 + e < n) out[base + (size_t)e] = res[e];
    }
  }
}

extern "C" void kernel_launch(void* const* d_in, const int* in_sizes, int n_in,
                              void* d_out, int out_size, void* d_ws, size_t ws_size,
                              hipStream_t stream) {
  (void)n_in; (void)out_size; (void)d_ws; (void)ws_size;
  const float* mask = (const float*)d_in[0];
  const float* adj  = (const float*)d_in[1];
  const int*   cid  = (const int*)d_in[2];
  float*       out  = (float*)d_out;
  const int n = in_sizes[2];  // cluster_id length == N (2708)

  dim3 grid((unsigned)((n + TS - 1) / TS), (unsigned)((n + TS - 1) / TS));
  mask_adj_kernel<<<grid, 256, 0, stream>>>(mask, adj, cid, out, n);
}